// ProbabilisticFractionalLayer_12463995093154
// MI455X (gfx1250) — compile-verified
//
#include <hip/hip_runtime.h>

typedef float v2f __attribute__((ext_vector_type(2)));
typedef float v8f __attribute__((ext_vector_type(8)));

#define T_LEN   4096
#define K_TAPS  256
#define NBLK    17            // Toeplitz K-blocks covering 256 taps with 16x16 tiles
#define WIN     512           // x window per row: 256 history + 256 fresh positions
#define XSTRIDE 516           // +4 pad -> LDS bank spread (4m+q distinct banks)
#define NFRAG   (NBLK * 4 * 2 * 16)   // 2176 packed float2 B-fragment entries
#define NLOAD4  (16 * (WIN / 4))      // 2048 float4 staging transfers per block

// ---------------------------------------------------------------------------
// Kernel 1: GL weights (sequential cumprod) + Toeplitz B-fragments packed in
// WMMA lane layout:  idx = ((bt*4+g)*2+kh)*16 + n
//   .x = w[n - (4g+2kh)   + 16*bt]   (VGPR0 holds K=2*kh of the 4-wide group)
//   .y = w[n - (4g+2kh+1) + 16*bt]   (VGPR1 holds K=2*kh+1)
// ---------------------------------------------------------------------------
__global__ void gl_weights_pack_kernel(const float* __restrict__ loc,
                                       const float* __restrict__ scale,
                                       const float* __restrict__ eps,
                                       v2f* __restrict__ twp) {
    __shared__ float w[K_TAPS];
    const int tid = threadIdx.x;
    if (tid == 0) {
        float alpha = loc[0] + log1pf(expf(scale[0])) * eps[0];
        float acc = 1.0f;
        w[0] = acc;
        for (int j = 1; j < K_TAPS; ++j) {
            acc *= (float(j) - 1.0f - alpha) / float(j);
            w[j] = acc;
        }
    }
    __syncthreads();
    for (int idx = tid; idx < NFRAG; idx += blockDim.x) {
        int n  = idx & 15;
        int r  = idx >> 4;
        int kh = r & 1;  r >>= 1;
        int g  = r & 3;
        int bt = r >> 2;
        int k  = 4 * g + 2 * kh;
        int j0 = n - k + 16 * bt;
        int j1 = j0 - 1;
        v2f v;
        v.x = (j0 >= 0 && j0 < K_TAPS) ? w[j0] : 0.0f;
        v.y = (j1 >= 0 && j1 < K_TAPS) ? w[j1] : 0.0f;
        twp[idx] = v;
    }
}

// ---------------------------------------------------------------------------
// Kernel 2: causal depthwise conv as sum of 17 Toeplitz GEMM blocks.
// Block = 256 threads = 8 wave32. Workgroup output: 16 rows x 256 positions.
// Each wave computes TWO 16x16 output tiles (n0 and n0+128) sharing every
// Toeplitz B-fragment -> 136 v_wmma_f32_16x16x4_f32 per wave, 1.5 ds_load_b64
// per WMMA. x window staged via async global->LDS DMA (ASYNCcnt path).
// ---------------------------------------------------------------------------
__global__ __launch_bounds__(256)
void frac_conv_wmma_kernel(const float* __restrict__ x,
                           const v2f* __restrict__ twp,
                           float* __restrict__ out) {
    __shared__ float xs[16][XSTRIDE];   // 16 rows x 512 window (+pad) = 33 KB
    __shared__ v2f   tw[NFRAG];         // packed Toeplitz B-fragments = 17 KB

    const int tid     = threadIdx.x;
    const int t0      = blockIdx.x * 256;   // fresh-position base of this block
    const int rowbase = blockIdx.y * 16;    // flattened (b*c) row base

    // Stage packed weight fragments (block-wide broadcast reuse).
    for (int i = tid; i < NFRAG; i += 256) tw[i] = twp[i];

    // Stage x window: rows rowbase..+15, positions t0-256 .. t0+255.
    // Async global->LDS B128 transfers; left edge (gpos<0) zero-filled via DS.
    for (int i = tid; i < NLOAD4; i += 256) {
        int row  = i >> 7;                 // / (WIN/4)
        int c4   = i & 127;                // % (WIN/4)
        int gpos = t0 - 256 + c4 * 4;
        float* lp = &xs[row][c4 * 4];
        unsigned ldsoff = (unsigned)(uintptr_t)lp;   // low 32 bits = LDS offset
        if (gpos >= 0) {
            unsigned long long ga = (unsigned long long)(uintptr_t)
                (x + (size_t)(rowbase + row) * T_LEN + gpos);
            asm volatile("global_load_async_to_lds_b128 %0, %1, off"
                         :: "v"(ldsoff), "v"(ga) : "memory");
        } else {
            *(float4*)lp = make_float4(0.0f, 0.0f, 0.0f, 0.0f);
        }
    }
    asm volatile("s_wait_asynccnt 0x0" ::: "memory");
    __syncthreads();

    const int lane = tid & 31;
    const int wv   = tid >> 5;     // 0..7 : first of this wave's two tiles
    const int m    = lane & 15;    // A-matrix M (row) == B/D N (position) index
    const int kh   = lane >> 4;    // which K pair within a 4-wide group

    v8f c0 = {0.0f, 0.0f, 0.0f, 0.0f, 0.0f, 0.0f, 0.0f, 0.0f};
    v8f c1 = {0.0f, 0.0f, 0.0f, 0.0f, 0.0f, 0.0f, 0.0f, 0.0f};

    // window offset of A[m, k] for block bt:  q = 16*(wv + 16 - bt) + k
    // second tile (wv+8) is the same offset + 128 elements.
    #pragma unroll
    for (int bt = 0; bt < NBLK; ++bt) {
        const int qbase = 16 * (wv + 16 - bt) + 2 * kh;
        #pragma unroll
        for (int g = 0; g < 4; ++g) {
            v2f b  = tw[((bt * 4 + g) * 2 + kh) * 16 + m];          // shared B
            v2f a0 = *(const v2f*)(&xs[m][qbase + 4 * g]);          // tile wv
            v2f a1 = *(const v2f*)(&xs[m][qbase + 4 * g + 128]);    // tile wv+8
            c0 = __builtin_amdgcn_wmma_f32_16x16x4_f32(
                     false, a0, false, b, (short)0, c0, false, false);
            c1 = __builtin_amdgcn_wmma_f32_16x16x4_f32(
                     false, a1, false, b, (short)0, c1, false, false);
        }
    }

    // D layout: VGPR v -> M = v + 8*kh (row), N = lane&15 (position)
    const int n0 = t0 + 16 * wv;
    #pragma unroll
    for (int v = 0; v < 8; ++v) {
        int row = rowbase + v + 8 * kh;
        out[(size_t)row * T_LEN + n0 + m]       = c0[v];
        out[(size_t)row * T_LEN + n0 + 128 + m] = c1[v];
    }
}

// ---------------------------------------------------------------------------
extern "C" void kernel_launch(void* const* d_in, const int* in_sizes, int n_in,
                              void* d_out, int out_size, void* d_ws, size_t ws_size,
                              hipStream_t stream) {
    const float* x     = (const float*)d_in[0];
    const float* loc   = (const float*)d_in[1];
    const float* scale = (const float*)d_in[2];
    const float* eps   = (const float*)d_in[3];
    // d_in[4] == k (256) is baked in as K_TAPS at compile time.
    (void)in_sizes; (void)n_in; (void)out_size; (void)ws_size;

    v2f*   twp  = (v2f*)d_ws;       // 2176 * 8 B = 17408 B of scratch
    float* outp = (float*)d_out;

    gl_weights_pack_kernel<<<1, 256, 0, stream>>>(loc, scale, eps, twp);

    dim3 grid(T_LEN / 256, (32 * 64) / 16);   // 16 time tiles x 128 row groups
    frac_conv_wmma_kernel<<<grid, 256, 0, stream>>>(x, twp, outp);
}